// RIMModule_32152125178148
// MI455X (gfx1250) — compile-verified
//
#include <hip/hip_runtime.h>

typedef float v2f __attribute__((ext_vector_type(2)));
typedef float v8f __attribute__((ext_vector_type(8)));

constexpr int B_ = 64, S_ = 512, D_ = 512, A_ = 256, H_ = 512, K_ = 8;
constexpr int KP_ = 16;   // K padded to WMMA N dimension

// Workspace layout (in floats)
constexpr size_t KQ_OFF  = 0;                            // [B,16,D]  zero-padded W_k@q
constexpr size_t DH_OFF  = KQ_OFF  + (size_t)B_*KP_*D_;  // [B,K,H]   hidden @ W_dh
constexpr size_t SIM_OFF = DH_OFF  + (size_t)B_*K_*H_;   // [B,K,S]   logits (null logit == 0)
constexpr size_t ATT_OFF = SIM_OFF + (size_t)B_*K_*S_;   // [B,16,S]  zero-padded softmax weights
constexpr size_t T_OFF   = ATT_OFF + (size_t)B_*KP_*S_;  // [B,K,D]   attn @ X

// ---------------------------------------------------------------------------
// Kernel 1: per (b,k):  q = hidden @ W_q ;  kq = W_k @ q (padded) ;
//                       dhb = hidden @ W_dh
// ---------------------------------------------------------------------------
__global__ __launch_bounds__(256) void proj_kernel(const float* __restrict__ hidden,
                                                   const float* __restrict__ Wq,
                                                   const float* __restrict__ Wk,
                                                   const float* __restrict__ Wdh,
                                                   float* __restrict__ ws)
{
    const int bk = blockIdx.x, b = bk / K_, k = bk % K_;
    const int tid = threadIdx.x;
    __shared__ float hid[H_];
    __shared__ float qs[A_];

    const float* hptr = hidden + ((size_t)b*K_ + k)*H_;
    hid[tid]       = hptr[tid];
    hid[tid + 256] = hptr[tid + 256];
    __syncthreads();

    // q[a] = sum_h hid[h] * W_q[k,h,a]   (coalesced over a)
    {
        const float* w = Wq + (size_t)k*H_*A_ + tid;
        float acc = 0.f;
        for (int h = 0; h < H_; ++h) acc += hid[h] * w[(size_t)h*A_];
        qs[tid] = acc;
    }
    __syncthreads();

    // kq[d] = sum_a W_k[k,d,a] * q[a]; also zero the padding row k+8
    float* kq  = ws + KQ_OFF + ((size_t)b*KP_ + k)*D_;
    float* kqz = ws + KQ_OFF + ((size_t)b*KP_ + k + K_)*D_;
    for (int d = tid; d < D_; d += 256) {
        const float* w = Wk + ((size_t)k*D_ + d)*A_;
        float acc = 0.f;
        for (int a = 0; a < A_; ++a) acc += w[a] * qs[a];
        kq[d]  = acc;
        kqz[d] = 0.f;
    }

    // dhb[g] = sum_h hid[h] * W_dh[k,h,g]   (coalesced over g)
    float* dhb = ws + DH_OFF + ((size_t)b*K_ + k)*H_;
    for (int g = tid; g < H_; g += 256) {
        const float* w = Wdh + (size_t)k*H_*H_ + g;
        float acc = 0.f;
        for (int h = 0; h < H_; ++h) acc += hid[h] * w[(size_t)h*H_];
        dhb[g] = acc;
    }
}

// ---------------------------------------------------------------------------
// Kernel 2 (WMMA, pass 1 over X): sim[b,k,s] = X_b(512x512) @ KQpad_b^T(512x16)
// Each wave owns TWO 16-row tiles (independent accumulators) sharing B loads.
// Outer loop advances one cacheline (32 f32) per row; prefetch once per line.
// ---------------------------------------------------------------------------
__global__ __launch_bounds__(256) void sim_kernel(const float* __restrict__ X,
                                                  float* __restrict__ ws)
{
    const int b     = blockIdx.y;
    const int wave  = threadIdx.x >> 5;
    const int lane  = threadIdx.x & 31;
    const int t     = blockIdx.x * 8 + wave;          // 0..15
    const int m0a   = t * 32;                         // first s-row of tile A
    const int m0b   = m0a + 16;                       // first s-row of tile B
    const int idx   = lane & 15;                      // A row / B,C column
    const int khalf = (lane >> 4) * 2;                // K pair base for this half-wave

    const float* Xb    = X + (size_t)b * S_ * D_;
    const float* arow0 = Xb + (size_t)(m0a + idx) * D_;
    const float* arow1 = Xb + (size_t)(m0b + idx) * D_;
    const float* brow  = ws + KQ_OFF + ((size_t)b*KP_ + idx) * D_;  // padded: no mask

    v8f c0 = {}, c1 = {};
    for (int kk = 0; kk < D_; kk += 32) {             // one 128B line per row per trip
        __builtin_prefetch(arow0 + kk + 96, 0, 1);    // speculative: OOB is dropped
        __builtin_prefetch(arow1 + kk + 96, 0, 1);
#pragma unroll
        for (int u = 0; u < 32; u += 4) {
            v2f a0, a1, bm;
            bm.x = brow[kk + u + khalf];
            bm.y = brow[kk + u + khalf + 1];
            a0.x = arow0[kk + u + khalf];
            a0.y = arow0[kk + u + khalf + 1];
            a1.x = arow1[kk + u + khalf];
            a1.y = arow1[kk + u + khalf + 1];
            c0 = __builtin_amdgcn_wmma_f32_16x16x4_f32(false, a0, false, bm,
                                                       (short)0, c0, false, false);
            c1 = __builtin_amdgcn_wmma_f32_16x16x4_f32(false, a1, false, bm,
                                                       (short)0, c1, false, false);
        }
    }

    // C element (vgpr j, lane): row M = j + 8*(lane>>4) -> s ; col N = lane&15 -> k
    if (idx < K_) {
        float* simk = ws + SIM_OFF + ((size_t)b*K_ + idx)*S_;
        const int mhi = (lane >> 4) * 8;
        for (int j = 0; j < 8; ++j) {
            simk[m0a + mhi + j] = c0[j];
            simk[m0b + mhi + j] = c1[j];
        }
    }
}

// ---------------------------------------------------------------------------
// Kernel 3: softmax over 513 logits per (b,k); logit[512] == 0 exactly
// (x's appended row is zero), folded in analytically. Writes zero-padded attn.
// One wave per k, 8 waves per block, one block per b.
// ---------------------------------------------------------------------------
__global__ __launch_bounds__(256) void softmax_kernel(float* __restrict__ ws)
{
    const int b    = blockIdx.x;
    const int k    = threadIdx.x >> 5;
    const int lane = threadIdx.x & 31;
    const float* sim  = ws + SIM_OFF + ((size_t)b*K_  + k)*S_;
    float*       attn = ws + ATT_OFF + ((size_t)b*KP_ + k)*S_;
    float*       attz = ws + ATT_OFF + ((size_t)b*KP_ + k + K_)*S_;

    float mx = 0.0f;                                  // the null logit
    for (int s = lane; s < S_; s += 32) mx = fmaxf(mx, sim[s]);
    for (int off = 16; off; off >>= 1) mx = fmaxf(mx, __shfl_xor(mx, off, 32));

    float sum = 0.f;
    for (int s = lane; s < S_; s += 32) sum += __expf(sim[s] - mx);
    for (int off = 16; off; off >>= 1) sum += __shfl_xor(sum, off, 32);
    sum += __expf(0.0f - mx);                         // null element's weight

    const float inv = 1.0f / sum;
    for (int s = lane; s < S_; s += 32) {
        attn[s] = __expf(sim[s] - mx) * inv;
        attz[s] = 0.f;                                // padding row k+8
    }
}

// ---------------------------------------------------------------------------
// Kernel 4 (WMMA, pass 2 over X): t[b,:,:] = attn_pad(16x512) @ X_b(512x512)
// Each wave owns TWO 16-column tiles sharing the A (attn) loads.
// One lane-spread prefetch per outer trip covers the next 32 rows' segments.
// ---------------------------------------------------------------------------
__global__ __launch_bounds__(256) void attend_kernel(const float* __restrict__ X,
                                                     float* __restrict__ ws)
{
    const int b     = blockIdx.y;
    const int wave  = threadIdx.x >> 5;
    const int lane  = threadIdx.x & 31;
    const int t     = blockIdx.x * 8 + wave;          // 0..15
    const int n0a   = t * 32;                         // first d-column of tile A
    const int n0b   = n0a + 16;
    const int idx   = lane & 15;
    const int khalf = (lane >> 4) * 2;

    const float* Xb   = X + (size_t)b * S_ * D_;
    const float* arow = ws + ATT_OFF + ((size_t)b*KP_ + idx) * S_;  // padded: no mask
    const float* pf   = Xb + (size_t)lane * D_ + n0a;   // lane-spread row prefetch base

    v8f c0 = {}, c1 = {};
    for (int kk = 0; kk < S_; kk += 32) {
        // 32 lanes -> rows kk+32 .. kk+63, one 128B segment each, one instruction
        __builtin_prefetch(pf + (size_t)(kk + 32) * D_, 0, 1);
#pragma unroll
        for (int u = 0; u < 32; u += 4) {
            const float* xr0 = Xb + (size_t)(kk + u + khalf) * D_;
            v2f a, b0, b1;
            a.x  = arow[kk + u + khalf];
            a.y  = arow[kk + u + khalf + 1];
            b0.x = xr0[n0a + idx];
            b0.y = xr0[D_ + n0a + idx];
            b1.x = xr0[n0b + idx];
            b1.y = xr0[D_ + n0b + idx];
            c0 = __builtin_amdgcn_wmma_f32_16x16x4_f32(false, a, false, b0,
                                                       (short)0, c0, false, false);
            c1 = __builtin_amdgcn_wmma_f32_16x16x4_f32(false, a, false, b1,
                                                       (short)0, c1, false, false);
        }
    }

    // lanes 0..15 hold rows M=j (valid k = 0..7); lanes 16..31 hold M=j+8 (padding)
    if (lane < 16) {
        float* tb = ws + T_OFF + (size_t)b * K_ * D_;
        for (int j = 0; j < K_; ++j) {
            tb[(size_t)j * D_ + n0a + lane] = c0[j];
            tb[(size_t)j * D_ + n0b + lane] = c1[j];
        }
    }
}

// ---------------------------------------------------------------------------
// Kernel 5: per (b,k): attended = t @ W_v ; cand = tanh(attended @ W_dx + dhb);
// out = mask ? cand : hidden, with mask = (k < active_kernels)  (null scores are
// all exactly 0, so stable top-k selects the lowest indices).
// ---------------------------------------------------------------------------
__global__ __launch_bounds__(256) void output_kernel(const float* __restrict__ hidden,
                                                     const float* __restrict__ Wv,
                                                     const float* __restrict__ Wdx,
                                                     const int* __restrict__ active,
                                                     const float* __restrict__ ws,
                                                     float* __restrict__ out)
{
    const int bk = blockIdx.x, b = bk / K_, k = bk % K_;
    const int tid = threadIdx.x;
    __shared__ float ts[D_];
    __shared__ float att[A_];

    const float* tb = ws + T_OFF + ((size_t)b*K_ + k)*D_;
    ts[tid]       = tb[tid];
    ts[tid + 256] = tb[tid + 256];
    __syncthreads();

    // attended[a] = sum_d t[d] * W_v[k,d,a]   (coalesced over a)
    {
        const float* w = Wv + (size_t)k*D_*A_ + tid;
        float acc = 0.f;
        for (int d = 0; d < D_; ++d) acc += ts[d] * w[(size_t)d*A_];
        att[tid] = acc;
    }
    __syncthreads();

    const float m = (k < active[0]) ? 1.0f : 0.0f;
    const float* dhb  = ws + DH_OFF + ((size_t)b*K_ + k)*H_;
    const float* hptr = hidden + ((size_t)b*K_ + k)*H_;
    float*       optr = out + ((size_t)b*K_ + k)*H_;
    for (int h = tid; h < H_; h += 256) {
        const float* w = Wdx + (size_t)k*A_*H_ + h;
        float acc = dhb[h];
        for (int a = 0; a < A_; ++a) acc += att[a] * w[(size_t)a*H_];
        const float cand = tanhf(acc);
        optr[h] = m * cand + (1.f - m) * hptr[h];
    }
}

// ---------------------------------------------------------------------------
extern "C" void kernel_launch(void* const* d_in, const int* in_sizes, int n_in,
                              void* d_out, int out_size, void* d_ws, size_t ws_size,
                              hipStream_t stream)
{
    const float* input  = (const float*)d_in[0];   // (B,S,D)
    const float* hidden = (const float*)d_in[1];   // (B,K,H)
    const float* Wq     = (const float*)d_in[2];   // (K,H,A)
    const float* Wk     = (const float*)d_in[3];   // (K,D,A)
    const float* Wv     = (const float*)d_in[4];   // (K,D,A)
    const float* Wdx    = (const float*)d_in[5];   // (K,A,H)
    const float* Wdh    = (const float*)d_in[6];   // (K,H,H)
    const int*   active = (const int*)d_in[7];     // scalar
    float* ws  = (float*)d_ws;
    float* out = (float*)d_out;                    // (B,K,H)

    proj_kernel   <<<B_*K_,        256, 0, stream>>>(hidden, Wq, Wk, Wdh, ws);
    sim_kernel    <<<dim3(2, B_),  256, 0, stream>>>(input, ws);
    softmax_kernel<<<B_,           256, 0, stream>>>(ws);
    attend_kernel <<<dim3(2, B_),  256, 0, stream>>>(input, ws);
    output_kernel <<<B_*K_,        256, 0, stream>>>(hidden, Wv, Wdx, active, ws, out);
}